// ClementsBellNxN_56358560858295
// MI455X (gfx1250) — compile-verified
//
#include <hip/hip_runtime.h>
#include <math.h>

// ---------------------------------------------------------------------------
// ClementsBellNxN on MI455X (gfx1250): fused 2x2 mixes -> block-diagonal 16x16
// complex tile products done with V_WMMA_F32_16X16X4_F32, matrix resident in
// LDS per 16-column slab.
// ---------------------------------------------------------------------------

typedef float v2f __attribute__((ext_vector_type(2)));
typedef float v8f __attribute__((ext_vector_type(8)));

__device__ __forceinline__ v8f wmma4(v2f a, v2f b, v8f c) {
  // D = A(16x4 f32) * B(4x16 f32) + C(16x16 f32)
  return __builtin_amdgcn_wmma_f32_16x16x4_f32(false, a, false, b, (short)0, c,
                                               false, false);
}

#define N_ROWS   512
#define LDS_ROWS 513          // +1 dummy row for odd-pass tail band
#define CT       16           // columns per block == WMMA N
#define NSTEPS   256

// Coupler constants: A = sqrt(0.95*0.505), B = sqrt(0.95*0.495)
#define C_A2 0.47975000f      // A^2 = 0.95*0.505
#define C_B2 0.47025000f      // B^2 = 0.95*0.495
#define C_AB 0.47497625f      // A*B = 0.95*sqrt(0.249975)

// One band: rows [rb, rb+16) of the LDS slab get  M <- T * M  where T is the
// 16x16 complex tile made of eight 2x2 pair blocks stored in table T
// (8 floats per pair: T00r,T00i,T01r,T01i, T01r,T01i,T11r,T11i).
__device__ __forceinline__ void band_update(float* sMr, float* sMi,
                                            const float* T, int rb, int pbase,
                                            int n /*lane&15*/, int kh /*lane>>4*/) {
  const int m      = n;        // A-matrix row index for this lane
  const int myPair = m >> 1;

  // A-operand source values for this lane (its pair's row of T)
  const float* tp = T + (size_t)(pbase + myPair) * 8 + (m & 1) * 4;
  const float c0r = tp[0], c0i = tp[1], c1r = tp[2], c1i = tp[3];

  // Load all 16 rows (real & imag) of the B operand before any writeback.
  float br[8], bim[8];
#pragma unroll
  for (int s4 = 0; s4 < 4; ++s4) {
    const int r0 = rb + 4 * s4 + 2 * kh;
    br[2 * s4 + 0]  = sMr[(r0 + 0) * CT + n];
    br[2 * s4 + 1]  = sMr[(r0 + 1) * CT + n];
    bim[2 * s4 + 0] = sMi[(r0 + 0) * CT + n];
    bim[2 * s4 + 1] = sMi[(r0 + 1) * CT + n];
  }

  v8f dr = {0.f, 0.f, 0.f, 0.f, 0.f, 0.f, 0.f, 0.f};
  v8f di = {0.f, 0.f, 0.f, 0.f, 0.f, 0.f, 0.f, 0.f};

#pragma unroll
  for (int s4 = 0; s4 < 4; ++s4) {
    // K-slice k0 = 4*s4 covers pair q = 2*s4 + kh for this lane half.
    const float act = (myPair == (2 * s4 + kh)) ? 1.0f : 0.0f;
    v2f ar  = {act * c0r, act * c1r};   //  Re(T) slice
    v2f ai  = {act * c0i, act * c1i};   //  Im(T) slice
    v2f ani = {-act * c0i, -act * c1i}; // -Im(T) slice
    v2f bR  = {br[2 * s4], br[2 * s4 + 1]};
    v2f bI  = {bim[2 * s4], bim[2 * s4 + 1]};
    dr = wmma4(ar, bR, dr);   // Dr += Tr*Mr
    dr = wmma4(ani, bI, dr);  // Dr -= Ti*Mi
    di = wmma4(ar, bI, di);   // Di += Tr*Mi
    di = wmma4(ai, bR, di);   // Di += Ti*Mr
  }

  // Writeback (C/D layout: VGPR v -> row rb + v + 8*kh, col n)
#pragma unroll
  for (int v = 0; v < 8; ++v) {
    const int r = rb + v + 8 * kh;
    sMr[r * CT + n] = dr[v];
    sMi[r * CT + n] = di[v];
  }
}

__global__ __launch_bounds__(256) void clements_wmma_kernel(
    const float* __restrict__ ph, float* __restrict__ out) {
  const int tid  = threadIdx.x;
  const int bx   = blockIdx.x;   // 16-column slab index (0..31)
  const int lane = tid & 31;
  const int wv   = tid >> 5;     // wave 0..7
  const int n    = lane & 15;
  const int kh   = lane >> 4;

  extern __shared__ float smem[];
  float* sMr = smem;                         // 513*16
  float* sMi = sMr + LDS_ROWS * CT;          // 513*16
  float* sTe = sMi + LDS_ROWS * CT;          // 256*8 even-pair table
  float* sTo = sTe + 256 * 8;                // 256*8 odd-pair table

  // ---- init: M = diag(exp(i*phases[0])), restricted to cols [16bx,16bx+16)
  for (int e = tid; e < LDS_ROWS * CT; e += 256) {
    sMr[e] = 0.0f;
    sMi[e] = 0.0f;
  }
  __syncthreads();
  if (tid < CT) {
    const int r = 16 * bx + tid;
    float s, c;
    sincosf(ph[r], &s, &c);
    sMr[r * CT + tid] = c;
    sMi[r * CT + tid] = s;
  }

#pragma unroll 1
  for (int st = 0; st < NSTEPS; ++st) {
    const float* pa = ph + (size_t)(1 + 2 * st) * N_ROWS;
    const float* pb = ph + (size_t)(2 + 2 * st) * N_ROWS;

    // ---- build fused 2x2 pair tables for this step (one pair per thread)
    {
      const int t = tid;
      float s0, c0, s1, c1;
      // even pair t: rows (2t, 2t+1), phases pa
      sincosf(pa[2 * t + 0], &s0, &c0);
      sincosf(pa[2 * t + 1], &s1, &c1);
      float* E = sTe + t * 8;
      E[0] = C_A2 * c0 - C_B2 * c1;  E[1] = C_A2 * s0 - C_B2 * s1;  // T00
      E[2] = -C_AB * (s0 + s1);      E[3] = C_AB * (c0 + c1);       // T01
      E[4] = E[2];                   E[5] = E[3];                   // T10
      E[6] = C_A2 * c1 - C_B2 * c0;  E[7] = C_A2 * s1 - C_B2 * s0;  // T11
      float* O = sTo + t * 8;
      if (t < 255) {
        // odd pair t: rows (2t+1, 2t+2), phases pb
        sincosf(pb[2 * t + 1], &s0, &c0);
        sincosf(pb[2 * t + 2], &s1, &c1);
        O[0] = C_A2 * c0 - C_B2 * c1;  O[1] = C_A2 * s0 - C_B2 * s1;
        O[2] = -C_AB * (s0 + s1);      O[3] = C_AB * (c0 + c1);
        O[4] = O[2];                   O[5] = O[3];
        O[6] = C_A2 * c1 - C_B2 * c0;  O[7] = C_A2 * s1 - C_B2 * s0;
      } else {
        // tail block: row 511 gets only phase pb[511]; row 512 is dummy (id)
        sincosf(pb[511], &s0, &c0);
        O[0] = c0;  O[1] = s0;
        O[2] = 0.f; O[3] = 0.f; O[4] = 0.f; O[5] = 0.f;
        O[6] = 1.f; O[7] = 0.f;
      }
    }
    __syncthreads();

    // ---- even pass: 32 aligned bands, 4 per wave
#pragma unroll
    for (int bi = 0; bi < 4; ++bi) {
      const int b = wv * 4 + bi;
      band_update(sMr, sMi, sTe, 16 * b, 8 * b, n, kh);
    }
    __syncthreads();

    // ---- odd pass: 32 bands offset by +1 row (last uses dummy row 512)
#pragma unroll
    for (int bi = 0; bi < 4; ++bi) {
      const int b = wv * 4 + bi;
      band_update(sMr, sMi, sTo, 16 * b + 1, 8 * b, n, kh);
    }
    // row 0 only gets the phase pb[0] (untouched by odd bands)
    if (tid < CT) {
      float s, c;
      sincosf(pb[0], &s, &c);
      const float xr = sMr[tid], xi = sMi[tid];
      sMr[tid] = xr * c - xi * s;
      sMi[tid] = xr * s + xi * c;
    }
    __syncthreads();
  }

  // ---- final per-row phase + store (complex64 interleaved re,im)
  for (int e = tid; e < N_ROWS * CT; e += 256) {
    const int r = e >> 4;
    const int c = e & 15;
    float s, cs;
    sincosf(ph[(size_t)(N_ROWS + 1) * N_ROWS + r], &s, &cs);
    const float xr = sMr[r * CT + c], xi = sMi[r * CT + c];
    const size_t o = ((size_t)r * N_ROWS + 16 * bx + c) * 2;
    out[o + 0] = xr * cs - xi * s;
    out[o + 1] = xr * s + xi * cs;
  }
}

extern "C" void kernel_launch(void* const* d_in, const int* in_sizes, int n_in,
                              void* d_out, int out_size, void* d_ws,
                              size_t ws_size, hipStream_t stream) {
  (void)in_sizes; (void)n_in; (void)d_ws; (void)ws_size; (void)out_size;
  const float* ph = (const float*)d_in[0];   // (514, 512) float32
  float* out = (float*)d_out;                // 512*512 complex64 -> float pairs
  const size_t lds_bytes =
      (size_t)(2 * LDS_ROWS * CT + 2 * 256 * 8) * sizeof(float);  // 82,048 B
  clements_wmma_kernel<<<dim3(N_ROWS / CT), dim3(256), lds_bytes, stream>>>(ph, out);
}